// PQBase_9706626089428
// MI455X (gfx1250) — compile-verified
//
#include <hip/hip_runtime.h>

#define S_DIM 16
#define K_DIM 256
#define D_CW  32
#define N_TOK 8192           // 4*2048 tokens
#define NT    128            // tokens per block (4 waves)
#define KC    64             // K chunk size
#define NCH   (K_DIM / KC)   // 4 chunks
#define RS    66             // padded LDS row stride for R (avoids bank conflicts)
#define EPSF  1e-5f
#define N_ELEM ((size_t)N_TOK * S_DIM * D_CW)   // 4,194,304
#define WCHUNK_B (KC * D_CW * 4)                // 8192 bytes per weight chunk

typedef float v2f __attribute__((ext_vector_type(2)));
typedef float v8f __attribute__((ext_vector_type(8)));

__device__ __forceinline__ void async_b128_to_lds(unsigned lds_off, unsigned gbyte_off,
                                                  unsigned long long gbase)
{
    // GLOBAL_LOAD_ASYNC_TO_LDS_B128 (GVS mode): copies 16 bytes from global base saddr plus byte offset vaddr to the LDS address in vdst
    asm volatile("global_load_async_to_lds_b128 %0, %1, %2 offset:0"
                 :: "v"(lds_off), "v"(gbyte_off), "s"(gbase) : "memory");
}

__device__ __forceinline__ void wait_async_lds()
{
    asm volatile("s_wait_asynccnt 0" ::: "memory");
}

__global__ __launch_bounds__(NT)
void pq_main(const float* __restrict__ z, const float* __restrict__ weight,
             float* __restrict__ out, float* __restrict__ ws)
{
    __shared__ __align__(16) float Wl[2][KC * D_CW];   // 16 KB double-buffered weight chunk
    __shared__ __align__(16) float Rl[NT * RS];        // 33 KB reciprocal distances
    __shared__ float sumR[NT];
    __shared__ int   aminS[NT];
    __shared__ float redQ[NT];
    __shared__ float redW[NT];

    const int tid  = threadIdx.x;
    const int lane = tid & 31;
    const int wv   = tid >> 5;
    const int s    = blockIdx.y;
    const int t    = blockIdx.x * NT + tid;     // this lane's token

    // Load this token's 32-dim sub-vector into VGPRs.
    const float* zp = z + (size_t)t * (S_DIM * D_CW) + s * D_CW;
    float zv[32];
    #pragma unroll
    for (int j = 0; j < 8; ++j) {
        float4 v = ((const float4*)zp)[j];
        zv[4*j+0] = v.x; zv[4*j+1] = v.y; zv[4*j+2] = v.z; zv[4*j+3] = v.w;
    }

    float dmin = 3.4e38f;
    int   kmin = 0;
    float sacc = 0.f;
    v8f c0 = {}, c1 = {}, c2 = {}, c3 = {};     // 4 C tiles per wave

    const float* wbase = weight + (size_t)s * K_DIM * D_CW;
    const int half = lane >> 4;
    const int lan  = lane & 15;

    const unsigned wl0 = (unsigned)(uintptr_t)&Wl[0][0];   // LDS byte offset of Wl

    // ---- prologue: async-stage chunk 0 into buffer 0 ----
    {
        const unsigned long long g = (unsigned long long)(uintptr_t)wbase;
        #pragma unroll
        for (int i = 0; i < WCHUNK_B / 16 / NT; ++i) {      // 4 x b128 per thread
            unsigned boff = (unsigned)((tid + i * NT) * 16);
            async_b128_to_lds(wl0 + boff, boff, g);
        }
    }
    wait_async_lds();
    __syncthreads();

    for (int ch = 0; ch < NCH; ++ch) {
        const int buf = ch & 1;

        // ---- async-stage next chunk into the other buffer (overlaps compute) ----
        if (ch + 1 < NCH) {
            const unsigned long long g =
                (unsigned long long)(uintptr_t)(wbase + (size_t)(ch + 1) * KC * D_CW);
            #pragma unroll
            for (int i = 0; i < WCHUNK_B / 16 / NT; ++i) {
                unsigned boff = (unsigned)((tid + i * NT) * 16);
                async_b128_to_lds(wl0 + (unsigned)((buf ^ 1) * WCHUNK_B) + boff, boff, g);
            }
        }

        // ---- Phase A: L1 distances + reciprocal weights (VALU) ----
        const float* Wc = &Wl[buf][0];
        for (int kk = 0; kk < KC; ++kk) {
            const float4* wr = (const float4*)(&Wc[kk * D_CW]);
            float dist = 0.f;
            #pragma unroll
            for (int j = 0; j < 8; ++j) {
                float4 w = wr[j];
                dist += fabsf(zv[4*j+0] - w.x);
                dist += fabsf(zv[4*j+1] - w.y);
                dist += fabsf(zv[4*j+2] - w.z);
                dist += fabsf(zv[4*j+3] - w.w);
            }
            const int k = ch * KC + kk;
            if (dist < dmin) { dmin = dist; kmin = k; }
            float rr = 1.0f / fmaxf(dist, EPSF);    // softmax(-log d) == (1/d)/sum(1/d)
            sacc += rr;
            Rl[tid * RS + kk] = rr;
        }
        __syncthreads();

        // ---- Phase B: C += R(128xKC) * W(KCx32) via V_WMMA_F32_16X16X4_F32 ----
        #pragma unroll
        for (int ti = 0; ti < 4; ++ti) {
            const int tile = wv * 4 + ti;
            const int tm = tile & 7;        // 8 M tiles
            const int tn = tile >> 3;       // 2 N tiles
            v8f c = (ti == 0) ? c0 : (ti == 1) ? c1 : (ti == 2) ? c2 : c3;
            const int arow = tm * 16 + lan;
            #pragma unroll
            for (int kk2 = 0; kk2 < KC / 4; ++kk2) {
                const int kb = kk2 * 4 + half * 2;
                // A frag (16x4 f32): lanes 0-15 hold K={0,1}, lanes 16-31 K={2,3}
                v2f a = *(const v2f*)&Rl[arow * RS + kb];
                // B frag (4x16 f32): mirrored striping, N = lane&15
                v2f b;
                b.x = Wc[(kb + 0) * D_CW + tn * 16 + lan];
                b.y = Wc[(kb + 1) * D_CW + tn * 16 + lan];
                c = __builtin_amdgcn_wmma_f32_16x16x4_f32(false, a, false, b,
                                                          (short)0, c, false, false);
            }
            if (ti == 0) c0 = c; else if (ti == 1) c1 = c;
            else if (ti == 2) c2 = c; else c3 = c;
        }

        // next-chunk async copy must be complete in *all* threads before reuse
        wait_async_lds();
        __syncthreads();
    }

    sumR[tid]  = sacc;
    aminS[tid] = kmin;

    // ---- z_q gather (exact) + loss_q ----
    float lq = 0.f;
    {
        const float* wq = wbase + (size_t)kmin * D_CW;
        float* oq = out + (size_t)t * (S_DIM * D_CW) + s * D_CW;
        #pragma unroll
        for (int j = 0; j < 8; ++j) {
            float4 w = ((const float4*)wq)[j];
            ((float4*)oq)[j] = w;
            float d0 = zv[4*j+0] - w.x, d1 = zv[4*j+1] - w.y;
            float d2 = zv[4*j+2] - w.z, d3 = zv[4*j+3] - w.w;
            lq += d0*d0 + d1*d1 + d2*d2 + d3*d3;
        }
    }
    __syncthreads();

    // ---- loss_w from C fragments: z_w = C / sum_r ----
    float lw = 0.f;
    #pragma unroll
    for (int ti = 0; ti < 4; ++ti) {
        const int tile = wv * 4 + ti;
        const int tm = tile & 7;
        const int tn = tile >> 3;
        v8f c = (ti == 0) ? c0 : (ti == 1) ? c1 : (ti == 2) ? c2 : c3;
        #pragma unroll
        for (int v = 0; v < 8; ++v) {
            // C layout: VGPR v, lanes 0-15 -> M=v, lanes 16-31 -> M=v+8; N = lane&15
            const int row = tm * 16 + half * 8 + v;
            const int d   = tn * 16 + lan;
            float zw = c[v] / sumR[row];
            float zq = wbase[(size_t)aminS[row] * D_CW + d];
            float df = zw - zq;
            lw += df * df;
        }
    }

    // ---- block reduction of partial losses ----
    redQ[tid] = lq; redW[tid] = lw;
    __syncthreads();
    for (int off = NT / 2; off > 0; off >>= 1) {
        if (tid < off) { redQ[tid] += redQ[tid + off]; redW[tid] += redW[tid + off]; }
        __syncthreads();
    }
    if (tid == 0) {
        const int bid = blockIdx.y * gridDim.x + blockIdx.x;
        ws[2 * bid + 0] = redQ[0];
        ws[2 * bid + 1] = redW[0];
    }
}

__global__ void pq_loss(const float* __restrict__ ws, float* __restrict__ out, int nb)
{
    if (threadIdx.x == 0 && blockIdx.x == 0) {
        float sq = 0.f, sw = 0.f;
        for (int i = 0; i < nb; ++i) { sq += ws[2*i]; sw += ws[2*i+1]; }
        out[N_ELEM] = (sq + sw) * (1.0f / (float)N_ELEM);   // loss_q + loss_w (same element count)
    }
}

extern "C" void kernel_launch(void* const* d_in, const int* in_sizes, int n_in,
                              void* d_out, int out_size, void* d_ws, size_t ws_size,
                              hipStream_t stream)
{
    (void)in_sizes; (void)n_in; (void)out_size; (void)ws_size;
    const float* z      = (const float*)d_in[0];
    const float* weight = (const float*)d_in[1];
    float* out = (float*)d_out;
    float* ws  = (float*)d_ws;

    dim3 grid(N_TOK / NT, S_DIM);
    pq_main<<<grid, NT, 0, stream>>>(z, weight, out, ws);
    pq_loss<<<1, 32, 0, stream>>>(ws, out, (N_TOK / NT) * S_DIM);
}